// SpatialGNNEncoder_90537910599956
// MI455X (gfx1250) — compile-verified
//
#include <hip/hip_runtime.h>
#include <hip/hip_bf16.h>

#define N_NODES 100000
#define N_EDGES 1000000
#define IN_DIM  8
#define HID     64
#define N_LAYERS 3
#define LN_EPS  1e-5f
#define SLOPE   0.2f
#define LDW     68   // padded LDS stride for the 64x64 weight tile (kills bank conflicts)

typedef float v2f __attribute__((ext_vector_type(2)));
typedef float v8f __attribute__((ext_vector_type(8)));

// ---------------- input projection: h = x @ Wp^T + bp ----------------
__global__ void gnn_proj_kernel(const float* __restrict__ x, const float* __restrict__ Wp,
                                const float* __restrict__ bp, float* __restrict__ h) {
  __shared__ float wp_s[HID * IN_DIM];   // 512 floats
  __shared__ float bp_s[HID];
  int t = threadIdx.x;
  wp_s[t] = Wp[t];
  wp_s[t + 256] = Wp[t + 256];
  if (t < HID) bp_s[t] = bp[t];
  __syncthreads();
  int gid = blockIdx.x * 256 + t;       // (node, col) pair, 64 cols per node
  int node = gid >> 6;
  int c = gid & 63;
  const float* xr = x + node * IN_DIM;
  const float* wr = wp_s + c * IN_DIM;
  float acc = bp_s[c];
#pragma unroll
  for (int k = 0; k < IN_DIM; ++k) acc += xr[k] * wr[k];
  h[gid] = acc;
}

// ---------------- dense layer GEMM via V_WMMA_F32_16X16X4_F32 ----------------
// out = h @ W^T + b ; grid.y==0 -> (Wself, bself, h_self), grid.y==1 -> (Wneigh, bneigh, h_neigh)
// block = 256 threads = 8 waves; wave w computes C tile [r0+16*(w>>2) .. +15] x [16*(w&3) .. +15]
__global__ void gnn_gemm_kernel(const float* __restrict__ h,
                                const float* __restrict__ Wself, const float* __restrict__ bself,
                                const float* __restrict__ Wneigh, const float* __restrict__ bneigh,
                                float* __restrict__ h_self, float* __restrict__ h_neigh) {
  __shared__ float w_s[HID * LDW];
  __shared__ float b_s[HID];
  const float* W  = blockIdx.y ? Wneigh : Wself;
  const float* b  = blockIdx.y ? bneigh : bself;
  float*       out = blockIdx.y ? h_neigh : h_self;

  int t = threadIdx.x;
#pragma unroll
  for (int i = 0; i < 16; ++i) {                 // 4096 elems / 256 threads
    int idx = t + i * 256;
    w_s[(idx >> 6) * LDW + (idx & 63)] = W[idx];
  }
  if (t < HID) b_s[t] = b[t];
  __syncthreads();

  int wave = t >> 5, lane = t & 31;
  int r0 = blockIdx.x * 32 + (wave >> 2) * 16;   // 100000/32 = 3125 blocks, exact
  int n0 = (wave & 3) * 16;
  int lr = lane & 15;
  int koff = (lane < 16) ? 0 : 2;                // ISA A/B fragment layout (K split across half-waves)

  const float* arow = h   + (size_t)(r0 + lr) * HID;  // A: row r0+lr, f32 pairs
  const float* brow = w_s + (n0 + lr) * LDW;          // B[k][n] = W[n][k]

  v8f c = {};
#pragma unroll
  for (int kb = 0; kb < 16; ++kb) {
    int k = kb * 4 + koff;
    v2f a  = *(const v2f*)(arow + k);
    v2f bf = *(const v2f*)(brow + k);
    c = __builtin_amdgcn_wmma_f32_16x16x4_f32(false, a, false, bf, (short)0, c, false, false);
  }

  float bias = b_s[n0 + lr];
  int mbase = r0 + ((lane < 16) ? 0 : 8);        // C layout: VGPR i -> rows i / i+8
#pragma unroll
  for (int i = 0; i < 8; ++i) {
    out[(size_t)(mbase + i) * HID + n0 + lr] = c[i] + bias;
  }
}

// ---------------- edge phase: gather, gate, scatter-add ----------------
// one wave per edge; lane owns columns {2*lane, 2*lane+1}
__global__ void gnn_edge_kernel(const int* __restrict__ eidx, const float* __restrict__ ew,
                                const float* __restrict__ Wedge, const float* __restrict__ h_neigh,
                                float* __restrict__ agg) {
  int wave = threadIdx.x >> 5, lane = threadIdx.x & 31;
  int e = blockIdx.x * 8 + wave;                 // 1e6 / 8 = 125000 blocks, exact
  int row = eidx[e];
  int col = eidx[N_EDGES + e];
  float w = ew[e];
  int c2 = lane * 2;
  float g0 = 1.f / (1.f + __expf(-w * Wedge[c2]));
  float g1 = 1.f / (1.f + __expf(-w * Wedge[c2 + 1]));
  v2f m = *(const v2f*)(h_neigh + (size_t)col * HID + c2);
  float* dst = agg + (size_t)row * HID + c2;
  atomicAdd(dst,     m.x * g0);
  atomicAdd(dst + 1, m.y * g1);
}

// ---------------- LayerNorm + leaky-ReLU + residual ----------------
// one wave per node; lane owns columns {2*lane, 2*lane+1}; wave32 shuffle reduction
__global__ void gnn_ln_kernel(const float* __restrict__ h_self, const float* __restrict__ agg,
                              const float* __restrict__ gamma, const float* __restrict__ beta,
                              const float* __restrict__ h, float* __restrict__ dst) {
  int wave = threadIdx.x >> 5, lane = threadIdx.x & 31;
  int node = blockIdx.x * 8 + wave;              // 100000/8 = 12500 blocks, exact
  size_t base = (size_t)node * HID + lane * 2;
  v2f a = *(const v2f*)(h_self + base);
  v2f g = *(const v2f*)(agg + base);
  float v0 = a.x + g.x, v1 = a.y + g.y;
  float s = v0 + v1, ss = v0 * v0 + v1 * v1;
#pragma unroll
  for (int m = 16; m >= 1; m >>= 1) {
    s  += __shfl_xor(s,  m, 32);
    ss += __shfl_xor(ss, m, 32);
  }
  float mu  = s * (1.f / HID);
  float var = ss * (1.f / HID) - mu * mu;
  float inv = rsqrtf(var + LN_EPS);
  int c = lane * 2;
  float o0 = (v0 - mu) * inv * gamma[c]     + beta[c];
  float o1 = (v1 - mu) * inv * gamma[c + 1] + beta[c + 1];
  o0 = (o0 >= 0.f) ? o0 : SLOPE * o0;
  o1 = (o1 >= 0.f) ? o1 : SLOPE * o1;
  v2f hh = *(const v2f*)(h + base);
  v2f r; r.x = hh.x + o0; r.y = hh.y + o1;
  *(v2f*)(dst + base) = r;
}

extern "C" void kernel_launch(void* const* d_in, const int* in_sizes, int n_in,
                              void* d_out, int out_size, void* d_ws, size_t ws_size,
                              hipStream_t stream) {
  const float* x      = (const float*)d_in[0];
  const int*   eidx   = (const int*)  d_in[1];
  const float* ew     = (const float*)d_in[2];
  const float* Wp     = (const float*)d_in[3];
  const float* bp     = (const float*)d_in[4];
  const float* Wself  = (const float*)d_in[5];
  const float* bself  = (const float*)d_in[6];
  const float* Wneigh = (const float*)d_in[7];
  const float* bneigh = (const float*)d_in[8];
  const float* Wedge  = (const float*)d_in[9];
  const float* gamma  = (const float*)d_in[10];
  const float* beta   = (const float*)d_in[11];
  float* out = (float*)d_out;

  const size_t NH = (size_t)N_NODES * HID;       // 6.4e6 floats = 25.6 MB
  float* h       = (float*)d_ws;
  float* h_self  = h + NH;
  float* h_neigh = h_self + NH;
  float* agg     = h_neigh + NH;

  // input projection
  gnn_proj_kernel<<<(N_NODES * HID) / 256, 256, 0, stream>>>(x, Wp, bp, h);

  for (int l = 0; l < N_LAYERS; ++l) {
    const float* Ws = Wself  + (size_t)l * HID * HID;
    const float* bs = bself  + (size_t)l * HID;
    const float* Wn = Wneigh + (size_t)l * HID * HID;
    const float* bn = bneigh + (size_t)l * HID;
    const float* We = Wedge  + (size_t)l * HID;
    const float* gm = gamma  + (size_t)l * HID;
    const float* bt = beta   + (size_t)l * HID;

    gnn_gemm_kernel<<<dim3(N_NODES / 32, 2), 256, 0, stream>>>(
        h, Ws, bs, Wn, bn, h_self, h_neigh);

    hipMemsetAsync(agg, 0, NH * sizeof(float), stream);

    gnn_edge_kernel<<<N_EDGES / 8, 256, 0, stream>>>(eidx, ew, We, h_neigh, agg);

    float* dst = (l == N_LAYERS - 1) ? out : h;
    gnn_ln_kernel<<<N_NODES / 8, 256, 0, stream>>>(h_self, agg, gm, bt, h, dst);
  }
}